// LSTM_86380382257772
// MI455X (gfx1250) — compile-verified
//
#include <hip/hip_runtime.h>

typedef float v2f __attribute__((ext_vector_type(2)));
typedef float v8f __attribute__((ext_vector_type(8)));

#define HSZ 4096
#define ROWS 16          // rows per block (WMMA M)
#define KSPLIT 4         // waves per matrix (K split)
#define NTHREADS 1024    // 32 waves: 8 matrices x 4 K-quarters
#define KCHUNK (HSZ / KSPLIT)   // 1024 elements per wave

struct Params {
    const float* x;
    const float* h;
    const float* c;
    const float* w[8];   // w_ii, w_if, w_ig, w_io, w_hi, w_hf, w_hg, w_ho
    const float* b[4];   // b_i, b_f, b_g, b_o
    float* out;          // [h_t(4096) | h_t(4096) | c_t(4096)]
};

__global__ __launch_bounds__(NTHREADS) void lstm_gemv_wmma(Params p) {
    __shared__ float s_x[HSZ];
    __shared__ float s_h[HSZ];
    __shared__ float s_part[8 * KSPLIT][ROWS];

    const int tid = threadIdx.x;

    // Stage x and h into LDS (each thread copies one float4 of each)
    {
        const float4* gx = (const float4*)p.x;
        const float4* gh = (const float4*)p.h;
        float4* lx = (float4*)s_x;
        float4* lh = (float4*)s_h;
        for (int i = tid; i < HSZ / 4; i += NTHREADS) {
            lx[i] = gx[i];
            lh[i] = gh[i];
        }
    }
    __syncthreads();

    const int lane  = tid & 31;
    const int wid   = tid >> 5;        // 0..31
    const int mat   = wid >> 2;        // 0..7  -> which weight matrix
    const int kq    = wid & (KSPLIT - 1); // 0..3 -> which K quarter
    const int row0  = blockIdx.x * ROWS;

    const int laneM = lane & 15;       // row within the 16-row tile
    const int hi    = lane >> 4;       // 0: K=k,k+1  1: K=k+2,k+3

    const float* W    = p.w[mat];
    const float* vecs = (mat < 4) ? s_x : s_h;

    // A: lane covers W[row0+laneM][k + 2*hi .. +1], streamed along the row.
    const float* aptr = W + (size_t)(row0 + laneM) * HSZ + (size_t)kq * KCHUNK + hi * 2;
    // B: only lanes 0 and 16 feed column 0 of D; other lanes' values only
    // affect discarded columns 1..15, so no masking is required.
    const float* bptr = vecs + kq * KCHUNK + hi * 2;

    v8f acc = {0.f, 0.f, 0.f, 0.f, 0.f, 0.f, 0.f, 0.f};
#pragma unroll 16
    for (int k = 0; k < KCHUNK; k += 4) {
        v2f a = *(const v2f*)(aptr + k);
        v2f b = *(const v2f*)(bptr + k);
        // D = A(16x4 f32) * B(4x16 f32) + C ; column 0 holds the GEMV partials
        acc = __builtin_amdgcn_wmma_f32_16x16x4_f32(
            /*neg_a=*/false, a, /*neg_b=*/false, b,
            /*c_mod=*/(short)0, acc, /*reuse_a=*/false, /*reuse_b=*/false);
    }

    // Column N=0 of D: lane 0 holds rows row0..row0+7 in acc[0..7],
    // lane 16 holds rows row0+8..row0+15.
    if (lane == 0) {
#pragma unroll
        for (int j = 0; j < 8; ++j) s_part[wid][j] = acc[j];
    } else if (lane == 16) {
#pragma unroll
        for (int j = 0; j < 8; ++j) s_part[wid][8 + j] = acc[j];
    }
    __syncthreads();

    // 16 threads finish the reduction + gate math for this block's 16 rows.
    if (tid < ROWS) {
        const int row = row0 + tid;
        float z[4];
#pragma unroll
        for (int g = 0; g < 4; ++g) {
            float zi = 0.f, zh = 0.f;
#pragma unroll
            for (int q = 0; q < KSPLIT; ++q) {
                zi += s_part[g * KSPLIT + q][tid];         // input-side  (w_i?)
                zh += s_part[(g + 4) * KSPLIT + q][tid];   // hidden-side (w_h?)
            }
            z[g] = zi + zh + p.b[g][row];
        }
        const float it = 1.f / (1.f + __expf(-z[0]));
        const float ft = 1.f / (1.f + __expf(-z[1]));
        const float gt = 1.f / (1.f + __expf(-z[2]));  // reference uses sigmoid here
        const float ot = 1.f / (1.f + __expf(-z[3]));
        const float ct = ft * p.c[row] + it * gt;
        const float ht = ot * tanhf(ct);
        p.out[row]            = ht;
        p.out[HSZ + row]      = ht;
        p.out[2 * HSZ + row]  = ct;
    }
}

extern "C" void kernel_launch(void* const* d_in, const int* in_sizes, int n_in,
                              void* d_out, int out_size, void* d_ws, size_t ws_size,
                              hipStream_t stream) {
    (void)in_sizes; (void)n_in; (void)d_ws; (void)ws_size; (void)out_size;
    Params p;
    p.x = (const float*)d_in[0];
    p.h = (const float*)d_in[1];
    p.c = (const float*)d_in[2];
    for (int i = 0; i < 8; ++i) p.w[i] = (const float*)d_in[3 + i];
    for (int i = 0; i < 4; ++i) p.b[i] = (const float*)d_in[11 + i];
    p.out = (float*)d_out;

    lstm_gemv_wmma<<<HSZ / ROWS, NTHREADS, 0, stream>>>(p);
}